// block_ad_7086696038749
// MI455X (gfx1250) — compile-verified
//
#include <hip/hip_runtime.h>
#include <hip/hip_bf16.h>
#include <math.h>

typedef __attribute__((ext_vector_type(2))) float v2f;
typedef __attribute__((ext_vector_type(8))) float v8f;

#define B_    128
#define C_IN  1024
#define C_OUT 1024
#define HID   256
#define HWHW  1024   // 32*32 spatial elements per (b,c) row

// ---------------------------------------------------------------------------
// Kernel 1: global average pool.  One wave32 per (b,c) row of 1024 floats.
// 8 coalesced float4 loads per lane (32 lanes * 8 * 4 = 1024 elems), then a
// wave shuffle reduction.  This kernel carries the entire 512 MiB roofline.
// ---------------------------------------------------------------------------
__global__ void pool_kernel(const float* __restrict__ in, float* __restrict__ x) {
    const int wave = blockIdx.x * (blockDim.x >> 5) + (threadIdx.x >> 5);
    const int lane = threadIdx.x & 31;
    const float* p = in + (size_t)wave * HWHW;
    float s = 0.0f;
#pragma unroll
    for (int i = 0; i < 8; ++i) {
        float4 v = *(const float4*)(p + (size_t)(i * 32 + lane) * 4);
        s += v.x + v.y + v.z + v.w;
    }
#pragma unroll
    for (int off = 16; off >= 1; off >>= 1)
        s += __shfl_down(s, off, 32);
    if (lane == 0) x[wave] = s * (1.0f / (float)HWHW);
}

// ---------------------------------------------------------------------------
// Kernel 2: build per-batch-row swap mask from swap_idx.
// Duplicated indices are idempotent (all write 1), matching at[idx].set.
// ---------------------------------------------------------------------------
__global__ void mask_kernel(const int* __restrict__ swap_idx, int n_idx,
                            int* __restrict__ mask) {
    const int t = threadIdx.x;
    if (t < B_) mask[t] = 0;
    __syncthreads();
    for (int i = t; i < n_idx; i += blockDim.x)
        mask[swap_idx[i]] = 1;
}

// ---------------------------------------------------------------------------
// Kernel 3: fc1 = relu(x @ W1^T + b1), fp32 WMMA 16x16x4.
// One wave per 16x16 tile of y [128 x 256]  -> 8*16 = 128 tiles.
// A fragment (16x4): lanes 0-15 K={0,1}, lanes 16-31 K={2,3}, M = lane&15.
// B fragment (4x16): same indexing with N in place of M; B[k][n] = W1[n][k].
// C/D (16x16 f32): vgpr r -> M = r + 8*(lane>>4), N = lane&15.
// ---------------------------------------------------------------------------
__global__ void fc1_kernel(const float* __restrict__ x,
                           const float* __restrict__ w1,
                           const float* __restrict__ b1,
                           float* __restrict__ y) {
    const int wave = blockIdx.x * (blockDim.x >> 5) + (threadIdx.x >> 5);
    const int lane = threadIdx.x & 31;
    const int half = lane >> 4;
    const int l16  = lane & 15;
    const int tm = (wave >> 4) * 16;   // 16 column tiles of 16 across HID=256
    const int tn = (wave & 15) * 16;

    v8f c = {};
    const float* ap = x  + (size_t)(tm + l16) * C_IN + half * 2;
    const float* bp = w1 + (size_t)(tn + l16) * C_IN + half * 2;
    for (int k = 0; k < C_IN; k += 4) {
        v2f a = *(const v2f*)(ap + k);
        v2f b = *(const v2f*)(bp + k);
        c = __builtin_amdgcn_wmma_f32_16x16x4_f32(
                /*neg_a=*/false, a, /*neg_b=*/false, b,
                /*c_mod=*/(short)0, c, /*reuse_a=*/false, /*reuse_b=*/false);
    }

    const int col = tn + l16;
    const float bias = b1[col];
#pragma unroll
    for (int r = 0; r < 8; ++r) {
        const int row = tm + r + half * 8;
        const float v = c[r] + bias;
        y[row * HID + col] = v > 0.0f ? v : 0.0f;
    }
}

// ---------------------------------------------------------------------------
// Kernel 4: fc2 + gating.  One wave per 16x16 tile of [128 x 1024] -> 512
// tiles.  After the WMMA K-loop: g = y2 + b2 + noise; v1 = clip(1.2*sig-0.1);
// v2 = (g>0); output selects v2 on swapped batch rows (forward of the
// straight-through estimator).
// ---------------------------------------------------------------------------
__global__ void fc2_gate_kernel(const float* __restrict__ y,
                                const float* __restrict__ w2,
                                const float* __restrict__ b2,
                                const float* __restrict__ noise,
                                const int* __restrict__ mask,
                                float* __restrict__ out) {
    const int wave = blockIdx.x * (blockDim.x >> 5) + (threadIdx.x >> 5);
    const int lane = threadIdx.x & 31;
    const int half = lane >> 4;
    const int l16  = lane & 15;
    const int tm = (wave >> 6) * 16;   // 64 column tiles of 16 across C_OUT
    const int tn = (wave & 63) * 16;

    v8f c = {};
    const float* ap = y  + (size_t)(tm + l16) * HID + half * 2;
    const float* bp = w2 + (size_t)(tn + l16) * HID + half * 2;
    for (int k = 0; k < HID; k += 4) {
        v2f a = *(const v2f*)(ap + k);
        v2f b = *(const v2f*)(bp + k);
        c = __builtin_amdgcn_wmma_f32_16x16x4_f32(
                false, a, false, b, (short)0, c, false, false);
    }

    const int col = tn + l16;
    const float bias = b2[col];
#pragma unroll
    for (int r = 0; r < 8; ++r) {
        const int row = tm + r + half * 8;
        const float g = c[r] + bias + noise[row * C_OUT + col];
        float v1 = 1.2f * (1.0f / (1.0f + __expf(-g))) - 0.1f;
        v1 = fminf(fmaxf(v1, 0.0f), 1.0f);
        const float v2 = g > 0.0f ? 1.0f : 0.0f;
        out[row * C_OUT + col] = mask[row] ? v2 : v1;
    }
}

// ---------------------------------------------------------------------------
extern "C" void kernel_launch(void* const* d_in, const int* in_sizes, int n_in,
                              void* d_out, int out_size, void* d_ws, size_t ws_size,
                              hipStream_t stream) {
    const float* input  = (const float*)d_in[0];  // [B, C_IN, 32, 32]
    const float* fc1_w  = (const float*)d_in[1];  // [HID, C_IN]
    const float* fc1_b  = (const float*)d_in[2];  // [HID]
    const float* fc2_w  = (const float*)d_in[3];  // [C_OUT, HID]
    const float* fc2_b  = (const float*)d_in[4];  // [C_OUT]
    const float* noise  = (const float*)d_in[5];  // [B, C_OUT]
    const int*   swapi  = (const int*)d_in[6];    // [C_OUT/2]
    const int    n_idx  = in_sizes[6];
    float* out = (float*)d_out;                   // [B, C_OUT] (flattened)

    // Workspace carve-up (all rewritten every call; deterministic)
    float* x    = (float*)d_ws;                   // 128*1024 floats = 512 KB
    float* yact = x + (size_t)B_ * C_IN;          // 128*256  floats = 128 KB
    int*   mask = (int*)(yact + (size_t)B_ * HID);// 128 ints

    // 1) pool: one wave per (b,c) row; 8 waves / 256-thread block
    pool_kernel<<<(B_ * C_IN) / 8, 256, 0, stream>>>(input, x);

    // 2) swap mask (independent of pool; ordered before fc2_gate by stream)
    mask_kernel<<<1, 512, 0, stream>>>(swapi, n_idx, mask);

    // 3) fc1: 128 tiles, 8 waves per block -> 16 blocks
    fc1_kernel<<<16, 256, 0, stream>>>(x, fc1_w, fc1_b, yact);

    // 4) fc2 + gating: 512 tiles -> 64 blocks
    fc2_gate_kernel<<<64, 256, 0, stream>>>(yact, fc2_w, fc2_b, noise, mask, out);
}